// LexicalEncoder_13846974562842
// MI455X (gfx1250) — compile-verified
//
#include <hip/hip_runtime.h>
#include <stdint.h>

typedef __attribute__((ext_vector_type(2))) float v2f;
typedef __attribute__((ext_vector_type(4))) float v4f;
typedef __attribute__((ext_vector_type(8))) float v8f;

#define B_ROWS   8192
#define DDIM     1024
#define KCODE    4096
#define TM       64            // residual rows per block in the GEMM kernel
#define LDS_STR  1028          // 1024 + 4 pad: bank-conflict-free b64 reads

// ---------------------------------------------------------------------------
// Kernel 1: scores = resid @ codebook^T (fp32 WMMA 16x16x4), fused row argmax.
//   Block: 256 threads (8 waves). A-tile (TM x DDIM) staged in LDS.
//   Waves: 4-way M split (16 rows each) x 2-way N split (half of 4096 each).
//   Each wave processes two 16-wide N-chunks per pass (two accumulators).
// ---------------------------------------------------------------------------
__global__ void lex_score_argmax(const float* __restrict__ resid,
                                 const float* __restrict__ codebook,
                                 int* __restrict__ amax) {
    extern __shared__ float smem[];
    float* at = smem;                                        // TM x LDS_STR
    unsigned long long* best = (unsigned long long*)(smem + TM * LDS_STR);

    const int tid   = threadIdx.x;
    const int mbase = blockIdx.x * TM;

    if (tid < TM) best[tid] = 0ull;

    // Stage A tile: coalesced float4 global loads -> padded LDS rows.
    for (int idx = tid * 4; idx < TM * DDIM; idx += 256 * 4) {
        const int r = idx >> 10;          // / DDIM
        const int c = idx & (DDIM - 1);
        v4f v = *(const v4f*)&resid[(size_t)(mbase + r) * DDIM + c];
        *(v4f*)&at[r * LDS_STR + c] = v;
    }
    __syncthreads();

    const int wave  = tid >> 5;
    const int lane  = tid & 31;
    const int mlane = lane & 15;
    const int khalf = lane >> 4;          // 0: K={0,1}, 1: K={2,3}
    const int msub  = (wave & 3) * 16;    // M sub-tile within block
    const int npar  = wave >> 2;          // 0/1: which half of the 256 chunks

    const float* aRow = &at[(msub + mlane) * LDS_STR + 2 * khalf];

    float    bestv[8];
    unsigned besti[8];
#pragma unroll
    for (int i = 0; i < 8; ++i) { bestv[i] = -3.4e38f; besti[i] = 0u; }

    for (int pc = 0; pc < 64; ++pc) {
        const int c0 = npar * 128 + pc * 2;       // chunk pair (c0, c0+1)
        const int n0 = c0 * 16;
        const float* b0 = &codebook[(size_t)(n0 + mlane) * DDIM + 2 * khalf];
        const float* b1 = b0 + (size_t)16 * DDIM;

        v8f acc0 = {};
        v8f acc1 = {};
#pragma unroll 4
        for (int d = 0; d < DDIM; d += 4) {
            v2f a   = *(const v2f*)(aRow + d);   // ds_load_b64, conflict-free
            v2f bb0 = *(const v2f*)(b0 + d);     // global_load_b64 (L2 resident)
            v2f bb1 = *(const v2f*)(b1 + d);
            acc0 = __builtin_amdgcn_wmma_f32_16x16x4_f32(
                false, a, false, bb0, (short)0, acc0, false, false);
            acc1 = __builtin_amdgcn_wmma_f32_16x16x4_f32(
                false, a, false, bb1, (short)0, acc1, false, false);
        }

        // Running per-lane argmax. n increases monotonically per lane slot and
        // strict '>' keeps the first occurrence (jnp.argmax semantics).
#pragma unroll
        for (int i = 0; i < 8; ++i) {
            const float s0 = acc0[i];
            const float s1 = acc1[i];
            const unsigned nn0 = (unsigned)(n0 + mlane);
            const unsigned nn1 = nn0 + 16u;
            if (s0 > bestv[i]) { bestv[i] = s0; besti[i] = nn0; }
            if (s1 > bestv[i]) { bestv[i] = s1; besti[i] = nn1; }
        }
    }

    // Cross-lane / cross-wave merge: pack (monotonic f32 key, ~idx) into u64
    // and ds_max_u64. ~idx => smaller index wins on exact score ties.
#pragma unroll
    for (int i = 0; i < 8; ++i) {
        const int rloc = msub + ((lane < 16) ? i : i + 8);
        unsigned u   = __float_as_uint(bestv[i]);
        unsigned key = (u & 0x80000000u) ? ~u : (u | 0x80000000u);
        unsigned long long packed =
            ((unsigned long long)key << 32) | (unsigned long long)(~besti[i]);
        atomicMax(&best[rloc], packed);
    }
    __syncthreads();

    if (tid < TM) {
        const unsigned low = (unsigned)(best[tid] & 0xFFFFFFFFull);
        amax[mbase + tid] = (int)(~low);
    }
}

// ---------------------------------------------------------------------------
// Kernel 2: per-row norm -> active update -> index emit -> residual update.
// One block (256 threads) per row; 4 floats per thread.
// ---------------------------------------------------------------------------
__global__ void lex_update(float* __restrict__ resid,
                           const float* __restrict__ codebook,
                           const int* __restrict__ amax,
                           int* __restrict__ flags,
                           int* __restrict__ idx_out,
                           float* __restrict__ resid_out,
                           float decay, int step, int write_resid) {
    const int r   = blockIdx.x;
    const int tid = threadIdx.x;
    const int c   = tid * 4;

    float* row = &resid[(size_t)r * DDIM];
    v4f v = *(const v4f*)&row[c];
    float p = v.x * v.x + v.y * v.y + v.z * v.z + v.w * v.w;

#pragma unroll
    for (int off = 16; off > 0; off >>= 1) p += __shfl_xor(p, off);

    __shared__ float red[8];
    __shared__ int   s_active;
    if ((tid & 31) == 0) red[tid >> 5] = p;
    __syncthreads();

    if (tid == 0) {
        float s = 0.f;
#pragma unroll
        for (int i = 0; i < 8; ++i) s += red[i];
        int act = flags[r];
        act = act && (sqrtf(s) >= 1e-3f);   // norm checked BEFORE the update
        flags[r]    = act;
        s_active    = act;
        idx_out[r * 8 + step] = act ? amax[r] : -1;
    }
    __syncthreads();

    if (s_active) {
        const int idx = amax[r];
        const v4f cv = *(const v4f*)&codebook[(size_t)idx * DDIM + c];
        v = v - decay * cv;
        *(v4f*)&row[c] = v;
    }
    if (write_resid) {
        *(v4f*)&resid_out[(size_t)r * DDIM + c] = v;
    }
}

// ---------------------------------------------------------------------------
// Kernel 0: residual <- targets, flags <- 1 (re-run every launch: stateless).
// ---------------------------------------------------------------------------
__global__ void lex_init(const float* __restrict__ targets,
                         float* __restrict__ resid,
                         int* __restrict__ flags) {
    const int r   = blockIdx.x;
    const int tid = threadIdx.x;
    const size_t off = (size_t)r * DDIM + tid * 4;
    *(v4f*)&resid[off] = *(const v4f*)&targets[off];
    if (tid == 0) flags[r] = 1;
}

extern "C" void kernel_launch(void* const* d_in, const int* in_sizes, int n_in,
                              void* d_out, int out_size, void* d_ws, size_t ws_size,
                              hipStream_t stream) {
    (void)in_sizes; (void)n_in; (void)out_size; (void)ws_size;

    const float* targets  = (const float*)d_in[0];   // [8192, 1024] f32
    const float* codebook = (const float*)d_in[1];   // [4096, 1024] f32

    char*  ws    = (char*)d_ws;
    float* resid = (float*)ws;
    int*   flags = (int*)(ws + (size_t)B_ROWS * DDIM * sizeof(float));
    int*   amax  = flags + B_ROWS;

    int*   idx_out   = (int*)d_out;                          // [8192, 8] i32
    float* resid_out = (float*)d_out + (size_t)B_ROWS * 8;   // [8192,1024] f32

    lex_init<<<B_ROWS, 256, 0, stream>>>(targets, resid, flags);

    const size_t lds_bytes =
        (size_t)TM * LDS_STR * sizeof(float) + TM * sizeof(unsigned long long);

    float decay = 1.0f;
    for (int s = 0; s < 8; ++s) {
        lex_score_argmax<<<B_ROWS / TM, 256, lds_bytes, stream>>>(
            resid, codebook, amax);
        lex_update<<<B_ROWS, 256, 0, stream>>>(
            resid, codebook, amax, flags, idx_out, resid_out,
            decay, s, (s == 7) ? 1 : 0);
        decay *= 0.9f;
    }
}